// Graph_Rec_50766513439159
// MI455X (gfx1250) — compile-verified
//
#include <hip/hip_runtime.h>
#include <math.h>

typedef __attribute__((ext_vector_type(2))) float v2f;
typedef __attribute__((ext_vector_type(8))) float v8f;

#define GEPS 1e-12f

// ---------------------------------------------------------------------------
// float atomic max via int/uint ordering trick (dst initialized to -inf)
// ---------------------------------------------------------------------------
__device__ inline void atomicMaxF32(float* addr, float val) {
    if (val >= 0.0f) {
        atomicMax((int*)addr, __float_as_int(val));
    } else {
        atomicMin((unsigned int*)addr, (unsigned int)__float_as_int(val));
    }
}

// ---------------------------------------------------------------------------
// init: out[0] = h0, msg = 0, m = -inf, s = 0
// ---------------------------------------------------------------------------
__global__ void init_kernel(const float* __restrict__ h0, float* __restrict__ out0,
                            float* __restrict__ m, float* __restrict__ s,
                            float* __restrict__ msg, int n) {
    int i = blockIdx.x * blockDim.x + threadIdx.x;
    int nd = n * 64;
    if (i < nd) { out0[i] = h0[i]; msg[i] = 0.0f; }
    if (i < n)  { m[i] = -__builtin_inff(); s[i] = 0.0f; }
}

__global__ void zero_kernel(float* __restrict__ p, int n) {
    int i = blockIdx.x * blockDim.x + threadIdx.x;
    if (i < n) p[i] = 0.0f;
}

// ---------------------------------------------------------------------------
// edge logits + segment max: 16 lanes per edge, float4 loads, shfl reduce
// ---------------------------------------------------------------------------
__global__ void logits_kernel(const float* __restrict__ h_att, const float* __restrict__ t_att,
                              const int* __restrict__ erow, const int* __restrict__ ecol,
                              float* __restrict__ m, float* __restrict__ wE, int n_edges) {
    int gid  = blockIdx.x * blockDim.x + threadIdx.x;
    int edge = gid >> 4;
    int sub  = gid & 15;
    int e    = edge < n_edges ? edge : n_edges - 1;   // keep all lanes active for shfl
    int r = erow[e], c = ecol[e];
    const float4 a = *(const float4*)(h_att + (size_t)r * 64 + sub * 4);
    const float4 b = *(const float4*)(t_att + (size_t)c * 64 + sub * 4);
    float dot = a.x * b.x + a.y * b.y + a.z * b.z + a.w * b.w;
    dot += __shfl_xor(dot, 1, 32);
    dot += __shfl_xor(dot, 2, 32);
    dot += __shfl_xor(dot, 4, 32);
    dot += __shfl_xor(dot, 8, 32);
    if (edge < n_edges && sub == 0) {
        wE[edge] = dot;
        atomicMaxF32(&m[r], dot);
    }
}

// w[e] = exp(logit - m[row]); s[row] += w[e]
__global__ void expsum_kernel(float* __restrict__ wE, const float* __restrict__ m,
                              float* __restrict__ s, const int* __restrict__ erow, int n_edges) {
    int e = blockIdx.x * blockDim.x + threadIdx.x;
    if (e >= n_edges) return;
    int r = erow[e];
    float v = expf(wE[e] - m[r]);
    wE[e] = v;
    atomicAdd(&s[r], v);
}

// w[e] /= max(s[row], eps)
__global__ void wnorm_kernel(float* __restrict__ wE, const float* __restrict__ s,
                             const int* __restrict__ erow, int n_edges) {
    int e = blockIdx.x * blockDim.x + threadIdx.x;
    if (e >= n_edges) return;
    wE[e] = wE[e] / fmaxf(s[erow[e]], GEPS);
}

// ---------------------------------------------------------------------------
// SpMM: msg[row] += w[e] * h[col].  One wave per 32 consecutive (row-sorted)
// edges; lane covers 2 of the 64 dims.  Register accumulation with flush on
// row change keeps f32 atomics ~1/degree of the naive count.
// ---------------------------------------------------------------------------
__global__ void spmm_kernel(const float* __restrict__ wE, const float* __restrict__ h,
                            const int* __restrict__ erow, const int* __restrict__ ecol,
                            float* __restrict__ msg, int n_edges) {
    const int lane = threadIdx.x & 31;
    const int wave = blockIdx.x * (blockDim.x >> 5) + (threadIdx.x >> 5);
    int e0 = wave * 32;
    if (e0 >= n_edges) return;
    int e1 = min(e0 + 32, n_edges);
    int cur = erow[e0];
    float ax = 0.0f, ay = 0.0f;
    for (int e = e0; e < e1; ++e) {
        int r = erow[e];
        if (r != cur) {
            atomicAdd(&msg[(size_t)cur * 64 + lane * 2    ], ax);
            atomicAdd(&msg[(size_t)cur * 64 + lane * 2 + 1], ay);
            ax = 0.0f; ay = 0.0f; cur = r;
        }
        float we = wE[e];
        const float2 hv = *(const float2*)(h + (size_t)ecol[e] * 64 + lane * 2);
        ax = fmaf(we, hv.x, ax);
        ay = fmaf(we, hv.y, ay);
    }
    atomicAdd(&msg[(size_t)cur * 64 + lane * 2    ], ax);
    atomicAdd(&msg[(size_t)cur * 64 + lane * 2 + 1], ay);
}

// ---------------------------------------------------------------------------
// Dense layer via V_WMMA_F32_16X16X4_F32.
// One wave per 16-node tile.  add = hprev + msg fused into the A load.
// acc[jb] (jb = feature block of 16) accumulated over 16 K-blocks of 4.
// Epilogue: +bias, ReLU, L2 row-norm (half-wave shfl_xor reduce), store.
//
// A 16x4 f32 layout: lane<16 -> (M=lane, K={0,1}); lane>=16 -> (M=lane-16, K={2,3})
// B 4x16 f32 layout: lane<16 -> (K={0,1}, N=lane); lane>=16 -> (K={2,3}, N=lane-16)
// C/D 16x16 f32: VGPR v, lanes 0-15: (M=v, N=lane); lanes 16-31: (M=v+8, N=lane-16)
// ---------------------------------------------------------------------------
__global__ void dense_kernel(const float* __restrict__ hprev, const float* __restrict__ msg,
                             const float* __restrict__ W, const float* __restrict__ bias,
                             float* __restrict__ out, int n) {
    const int lane = threadIdx.x & 31;
    const int wave = threadIdx.x >> 5;
    const int tile = blockIdx.x * (blockDim.x >> 5) + wave;
    const int ntiles = (n + 15) >> 4;
    if (tile >= ntiles) return;

    const int half = lane >> 4;       // 0: K+0/1 (rows 0-7 in C), 1: K+2/3 (rows 8-15)
    const int l16  = lane & 15;
    int row  = tile * 16 + l16;
    int rowc = row < n ? row : n - 1; // clamp: all lanes must execute WMMA

    v8f acc[4] = {};

#pragma unroll
    for (int kb = 0; kb < 16; ++kb) {
        const int koff = kb * 4 + half * 2;
        const float* hp = hprev + (size_t)rowc * 64 + koff;
        const float* mp = msg   + (size_t)rowc * 64 + koff;
        v2f a;
        a.x = hp[0] + mp[0];
        a.y = hp[1] + mp[1];
#pragma unroll
        for (int jb = 0; jb < 4; ++jb) {
            const float* wp = W + (size_t)(jb * 16 + l16) * 64 + koff;
            v2f b;
            b.x = wp[0];
            b.y = wp[1];
            // D = A(16x4) * B(4x16) + C, full f32
            acc[jb] = __builtin_amdgcn_wmma_f32_16x16x4_f32(
                false, a, false, b, (short)0, acc[jb], false, false);
        }
    }

    float bj[4];
#pragma unroll
    for (int jb = 0; jb < 4; ++jb) bj[jb] = bias[jb * 16 + l16];

#pragma unroll
    for (int v = 0; v < 8; ++v) {
        float vals[4];
        float ss = 0.0f;
#pragma unroll
        for (int jb = 0; jb < 4; ++jb) {
            float y = fmaxf(acc[jb][v] + bj[jb], 0.0f);   // bias + ReLU
            vals[jb] = y;
            ss = fmaf(y, y, ss);
        }
        // reduce across the 16 lanes holding this row's 64 features
        ss += __shfl_xor(ss, 1, 32);
        ss += __shfl_xor(ss, 2, 32);
        ss += __shfl_xor(ss, 4, 32);
        ss += __shfl_xor(ss, 8, 32);
        float inv = 1.0f / fmaxf(sqrtf(ss), GEPS);
        int m    = v + half * 8;
        int orow = tile * 16 + m;
        if (orow < n) {
#pragma unroll
            for (int jb = 0; jb < 4; ++jb)
                out[(size_t)orow * 64 + jb * 16 + l16] = vals[jb] * inv;
        }
    }
}

// ---------------------------------------------------------------------------
// launch
// ---------------------------------------------------------------------------
static inline int cdiv(long long a, long long b) { return (int)((a + b - 1) / b); }

extern "C" void kernel_launch(void* const* d_in, const int* in_sizes, int n_in,
                              void* d_out, int out_size, void* d_ws, size_t ws_size,
                              hipStream_t stream) {
    const float* h0    = (const float*)d_in[0];
    const float* h_att = (const float*)d_in[1];
    const float* t_att = (const float*)d_in[2];
    const float* w1    = (const float*)d_in[3];
    const float* b1    = (const float*)d_in[4];
    const float* w2    = (const float*)d_in[5];
    const float* b2    = (const float*)d_in[6];
    const int*   erow  = (const int*)d_in[7];
    const int*   ecol  = (const int*)d_in[8];
    float* out = (float*)d_out;

    const int N = in_sizes[0] / 64;
    const int E = in_sizes[7];

    // workspace: m[N] | s[N] | wE[E] | msg[N*64]
    float* ws  = (float*)d_ws;
    float* m   = ws;
    float* s   = ws + N;
    float* wE  = ws + 2 * (size_t)N;
    float* msg = ws + 2 * (size_t)N + E;

    float* out1 = out + (size_t)N * 64;
    float* out2 = out + 2 * (size_t)N * 64;

    const int TPB = 256;
    const int nd  = N * 64;

    // edge softmax
    init_kernel<<<cdiv(nd, TPB), TPB, 0, stream>>>(h0, out, m, s, msg, N);
    logits_kernel<<<cdiv((long long)E * 16, TPB), TPB, 0, stream>>>(h_att, t_att, erow, ecol, m, wE, E);
    expsum_kernel<<<cdiv(E, TPB), TPB, 0, stream>>>(wE, m, s, erow, E);
    wnorm_kernel<<<cdiv(E, TPB), TPB, 0, stream>>>(wE, s, erow, E);

    const int waves   = cdiv(E, 32);
    const int spmm_bl = cdiv(waves, TPB / 32);
    const int ntiles  = (N + 15) >> 4;
    const int dense_bl = cdiv(ntiles, TPB / 32);

    // layer 1: h = h0
    spmm_kernel<<<spmm_bl, TPB, 0, stream>>>(wE, h0, erow, ecol, msg, E);
    dense_kernel<<<dense_bl, TPB, 0, stream>>>(h0, msg, w1, b1, out1, N);

    // layer 2: h = norm1
    zero_kernel<<<cdiv(nd, TPB), TPB, 0, stream>>>(msg, nd);
    spmm_kernel<<<spmm_bl, TPB, 0, stream>>>(wE, out1, erow, ecol, msg, E);
    dense_kernel<<<dense_bl, TPB, 0, stream>>>(out1, msg, w2, b2, out2, N);
}